// SSIM_75642964017466
// MI455X (gfx1250) — compile-verified
//
#include <hip/hip_runtime.h>

typedef float v2f __attribute__((ext_vector_type(2)));
typedef float v8f __attribute__((ext_vector_type(8)));

#define H_IMG 1024
#define W_IMG 2048
#define W_SLICE 1024
#define C1F 0.0001f
#define C2F 0.0009f
#define EPSF 1e-8f

// Normalized 11-tap Gaussian (sigma=1.5), zero-padded so that
// PADG[i + 15] == g[i] for i in [0,10], and 0 for i in [-15,-1] U [11,27].
__constant__ float PADG[43] = {
    0.f, 0.f, 0.f, 0.f, 0.f, 0.f, 0.f, 0.f, 0.f, 0.f,
    0.f, 0.f, 0.f, 0.f, 0.f,
    0.00102838f, 0.00759875f, 0.03600077f, 0.10936069f, 0.21300553f,
    0.26601173f,
    0.21300553f, 0.10936069f, 0.03600077f, 0.00759875f, 0.00102838f,
    0.f, 0.f, 0.f, 0.f, 0.f, 0.f, 0.f, 0.f, 0.f, 0.f,
    0.f, 0.f, 0.f, 0.f, 0.f, 0.f, 0.f
};

// ---------- monotone float <-> uint keys for atomic min/max ----------
__device__ __forceinline__ unsigned fkey(float f) {
    unsigned b = __float_as_uint(f);
    return (b & 0x80000000u) ? ~b : (b | 0x80000000u);
}
__device__ __forceinline__ float funkey(unsigned u) {
    unsigned b = (u & 0x80000000u) ? (u & 0x7fffffffu) : ~u;
    return __uint_as_float(b);
}

__device__ __forceinline__ v8f wmma4(v2f a, v2f b, v8f c) {
    return __builtin_amdgcn_wmma_f32_16x16x4_f32(
        false, a, false, b, (short)0, c, false, false);
}

// ---------------------------------------------------------------------
// Kernel 0: reset workspace accumulators (ws is poisoned by harness)
// ---------------------------------------------------------------------
__global__ void init_ws(unsigned* keys, double* acc) {
    int t = threadIdx.x;
    if (t < 32) {
        bool isMin = (t < 8) || (t >= 16 && t < 24);
        keys[t] = isMin ? 0xFFFFFFFFu : 0u;
    }
    if (t == 0) *acc = 0.0;
}

// ---------------------------------------------------------------------
// Kernel 1: per-batch min/max over the 1024x1024 column slice
// ---------------------------------------------------------------------
__global__ __launch_bounds__(256) void minmax_kernel(
    const float* __restrict__ img1, const float* __restrict__ img2,
    const int* __restrict__ mask_pos, unsigned* __restrict__ keys)
{
    const int b = blockIdx.y;
    const int chunk = blockIdx.x;        // 64 chunks of 16 rows
    const int tid = threadIdx.x;
    const int mp = mask_pos[b];
    const float* p1 = img1 + (size_t)b * H_IMG * W_IMG;
    const float* p2 = img2 + (size_t)b * H_IMG * W_IMG;

    float mn1 = 3.4e38f, mx1 = -3.4e38f, mn2 = 3.4e38f, mx2 = -3.4e38f;
    for (int t = tid; t < 16 * 1024; t += 256) {
        int r = chunk * 16 + (t >> 10);
        int c = t & 1023;
        size_t off = (size_t)r * W_IMG + mp + c;
        float v1 = p1[off], v2 = p2[off];
        mn1 = fminf(mn1, v1); mx1 = fmaxf(mx1, v1);
        mn2 = fminf(mn2, v2); mx2 = fmaxf(mx2, v2);
    }
#pragma unroll
    for (int off = 16; off > 0; off >>= 1) {
        mn1 = fminf(mn1, __shfl_xor(mn1, off, 32));
        mx1 = fmaxf(mx1, __shfl_xor(mx1, off, 32));
        mn2 = fminf(mn2, __shfl_xor(mn2, off, 32));
        mx2 = fmaxf(mx2, __shfl_xor(mx2, off, 32));
    }
    __shared__ float red[4][8];
    int wid = tid >> 5;
    if ((tid & 31) == 0) {
        red[0][wid] = mn1; red[1][wid] = mx1;
        red[2][wid] = mn2; red[3][wid] = mx2;
    }
    __syncthreads();
    if (tid == 0) {
#pragma unroll
        for (int w = 1; w < 8; ++w) {
            mn1 = fminf(mn1, red[0][w]); mx1 = fmaxf(mx1, red[1][w]);
            mn2 = fminf(mn2, red[2][w]); mx2 = fmaxf(mx2, red[3][w]);
        }
        atomicMin(&keys[0 * 8 + b], fkey(mn1));
        atomicMax(&keys[1 * 8 + b], fkey(mx1));
        atomicMin(&keys[2 * 8 + b], fkey(mn2));
        atomicMax(&keys[3 * 8 + b], fkey(mx2));
    }
}

// ---------------------------------------------------------------------
// Kernel 2: fused normalize + 5x separable 11x11 Gaussian conv (WMMA)
//           + SSIM map + reduction.  One wave32 per 16x16 output tile.
//           A-fragments loaded ONCE per K-chunk feed 5 interleaved
//           WMMA accumulator chains (mu1, mu2, E11, E22, E12).
// ---------------------------------------------------------------------
__global__ __launch_bounds__(32) void ssim_tile_kernel(
    const float* __restrict__ img1, const float* __restrict__ img2,
    const int* __restrict__ mask_pos, const unsigned* __restrict__ keys,
    double* __restrict__ acc)
{
    __shared__ float In1[32 * 28];      // 26x26 halo tile, padded to 32x28
    __shared__ float In2[32 * 28];
    __shared__ float Mid[5 * 32 * 16];  // horizontal-pass result, 5 planes

    const int lane = threadIdx.x;
    const int tx = blockIdx.x, ty = blockIdx.y, b = blockIdx.z;
    const int mp = mask_pos[b];

    const float mn1 = funkey(keys[0 * 8 + b]);
    const float mx1 = funkey(keys[1 * 8 + b]);
    const float mn2 = funkey(keys[2 * 8 + b]);
    const float mx2 = funkey(keys[3 * 8 + b]);
    const float inv1 = 1.f / (mx1 - mn1 + EPSF);
    const float inv2 = 1.f / (mx2 - mn2 + EPSF);

    const float* p1 = img1 + (size_t)b * H_IMG * W_IMG;
    const float* p2 = img2 + (size_t)b * H_IMG * W_IMG;

    const int R0 = ty * 16 - 5;   // halo origin (SAME padding, pad = 5)
    const int C0 = tx * 16 - 5;

    // ---- coalesced halo fill: one row per iteration, lane = column ----
    if (lane < 28) {
        const int c = C0 + lane;
        const bool cok = (lane < 26) && (c >= 0) && (c < W_SLICE);
        for (int t = 0; t < 32; ++t) {
            const int r = R0 + t;
            float v1 = 0.f, v2 = 0.f;
            if (cok && t < 26 && r >= 0 && r < H_IMG) {
                size_t off = (size_t)r * W_IMG + mp + c;
                v1 = (p1[off] - mn1) * inv1;
                v2 = (p2[off] - mn2) * inv2;
            }
            In1[t * 28 + lane] = v1;
            In2[t * 28 + lane] = v2;
        }
    }
    __syncthreads();

    const int n = lane & 15;
    const int khi = (lane >> 4) ? 2 : 0;      // K sub-offset for this half
    const int moff = (lane >> 4) ? 8 : 0;     // C-fragment M offset

    // Banded filter B-fragments: B[k][n] = g[k-n]; identical fragments are
    // reused by the transposed vertical pass (W^T[k][r] = g[k-r]).
    v2f bf[7];
#pragma unroll
    for (int kk = 0; kk < 7; ++kk) {
        int k0 = 4 * kk + khi;
        bf[kk].x = PADG[k0 - n + 15];
        bf[kk].y = PADG[k0 - n + 16];
    }

    // ---- horizontal pass: 2 M-tiles x 7 K-chunks, 5 maps per load ----
#pragma unroll
    for (int mt = 0; mt < 2; ++mt) {
        v8f a0 = {}, a1 = {}, a2 = {}, a3 = {}, a4 = {};
        const float* r1 = &In1[(mt * 16 + n) * 28];
        const float* r2 = &In2[(mt * 16 + n) * 28];
#pragma unroll
        for (int kk = 0; kk < 7; ++kk) {
            const int k0 = 4 * kk + khi;          // even -> 8B aligned
            v2f x = *(const v2f*)(r1 + k0);
            v2f y = *(const v2f*)(r2 + k0);
            v2f xx = x * x, yy = y * y, xy = x * y;
            a0 = wmma4(x,  bf[kk], a0);
            a1 = wmma4(y,  bf[kk], a1);
            a2 = wmma4(xx, bf[kk], a2);
            a3 = wmma4(yy, bf[kk], a3);
            a4 = wmma4(xy, bf[kk], a4);
        }
#pragma unroll
        for (int v = 0; v < 8; ++v) {
            const int k = mt * 16 + moff + v;
            Mid[0 * 512 + k * 16 + n] = a0[v];
            Mid[1 * 512 + k * 16 + n] = a1[v];
            Mid[2 * 512 + k * 16 + n] = a2[v];
            Mid[3 * 512 + k * 16 + n] = a3[v];
            Mid[4 * 512 + k * 16 + n] = a4[v];
        }
    }
    __syncthreads();

    // ---- vertical pass, transposed: Out^T = Mid^T x W^T ----
    v8f r0 = {}, r1v = {}, r2v = {}, r3v = {}, r4v = {};
#pragma unroll
    for (int kk = 0; kk < 7; ++kk) {
        const int k0 = 4 * kk + khi;
        v2f m0, m1, m2, m3, m4;
        m0.x = Mid[0 * 512 + k0 * 16 + n]; m0.y = Mid[0 * 512 + (k0 + 1) * 16 + n];
        m1.x = Mid[1 * 512 + k0 * 16 + n]; m1.y = Mid[1 * 512 + (k0 + 1) * 16 + n];
        m2.x = Mid[2 * 512 + k0 * 16 + n]; m2.y = Mid[2 * 512 + (k0 + 1) * 16 + n];
        m3.x = Mid[3 * 512 + k0 * 16 + n]; m3.y = Mid[3 * 512 + (k0 + 1) * 16 + n];
        m4.x = Mid[4 * 512 + k0 * 16 + n]; m4.y = Mid[4 * 512 + (k0 + 1) * 16 + n];
        r0  = wmma4(m0, bf[kk], r0);
        r1v = wmma4(m1, bf[kk], r1v);
        r2v = wmma4(m2, bf[kk], r2v);
        r3v = wmma4(m3, bf[kk], r3v);
        r4v = wmma4(m4, bf[kk], r4v);
    }

    // ---- per-pixel SSIM + tile reduction ----
    float partial = 0.f;
#pragma unroll
    for (int v = 0; v < 8; ++v) {
        float mu1 = r0[v], mu2 = r1v[v];
        float e11 = r2v[v], e22 = r3v[v], e12 = r4v[v];
        float mu1s = mu1 * mu1, mu2s = mu2 * mu2, mu12 = mu1 * mu2;
        float s1 = e11 - mu1s, s2 = e22 - mu2s, s12 = e12 - mu12;
        float num = (2.f * mu12 + C1F) * (2.f * s12 + C2F);
        float den = (mu1s + mu2s + C1F) * (s1 + s2 + C2F);
        partial += num / den;
    }
#pragma unroll
    for (int off = 16; off > 0; off >>= 1)
        partial += __shfl_xor(partial, off, 32);
    if (lane == 0) atomicAdd(acc, (double)partial);
}

// ---------------------------------------------------------------------
// Kernel 3: finalize mean
// ---------------------------------------------------------------------
__global__ void finalize_kernel(const double* __restrict__ acc,
                                float* __restrict__ out)
{
    if (threadIdx.x == 0 && blockIdx.x == 0)
        out[0] = (float)(acc[0] / (8.0 * 1024.0 * 1024.0));
}

// ---------------------------------------------------------------------
extern "C" void kernel_launch(void* const* d_in, const int* in_sizes, int n_in,
                              void* d_out, int out_size, void* d_ws, size_t ws_size,
                              hipStream_t stream)
{
    (void)in_sizes; (void)n_in; (void)out_size; (void)ws_size;
    const float* img1     = (const float*)d_in[0];
    const float* img2     = (const float*)d_in[1];
    const int*   mask_pos = (const int*)d_in[2];
    // d_in[3] = mask_width (scalar 1024; geometry hardcoded to match setup)

    unsigned* keys = (unsigned*)d_ws;                    // 32 u32 min/max keys
    double*   acc  = (double*)((char*)d_ws + 256);       // global SSIM sum
    float*    out  = (float*)d_out;

    init_ws<<<1, 64, 0, stream>>>(keys, acc);
    minmax_kernel<<<dim3(64, 8), 256, 0, stream>>>(img1, img2, mask_pos, keys);
    ssim_tile_kernel<<<dim3(64, 64, 8), 32, 0, stream>>>(img1, img2, mask_pos, keys, acc);
    finalize_kernel<<<1, 32, 0, stream>>>(acc, out);
}